// Model_53601191854545
// MI455X (gfx1250) — compile-verified
//
#include <hip/hip_runtime.h>

#define NP_PEP  50000
#define NP_PROT 8000
#define NE      1600000
#define NEL     200000
#define F_IN    1280
#define HD      128

typedef float v2f __attribute__((ext_vector_type(2)));
typedef float v8f __attribute__((ext_vector_type(8)));

// -----------------------------------------------------------------------------
// C[N,128] = A[N,K] @ W[K,128] (+ bias) (+ Cin) (+ ReLU)
// One wave per 16-row strip; 8 x (16x16) f32 accumulators cover all 128 cols.
// Uses V_WMMA_F32_16X16X4_F32 (wave32, K-step 4).
// A frag (16x4): lane l: row = l%16; VGPR0/1 = K(k + (l/16)*2 + {0,1})
// B frag (4x16): lane l: col = l%16; VGPR0/1 = K(k + (l/16)*2 + {0,1})
// C/D (16x16):   VGPR g: row = g + (l/16)*8, col = l%16
// -----------------------------------------------------------------------------
__global__ __launch_bounds__(256)
void gemm128_wmma(const float* __restrict__ A, int N, int K,
                  const float* __restrict__ W,
                  const float* __restrict__ bias,
                  const float* __restrict__ Cin,
                  float* __restrict__ Cout,
                  int relu)
{
    const int lane = threadIdx.x & 31;
    const int wave = threadIdx.x >> 5;
    const int row0 = (blockIdx.x * 8 + wave) * 16;
    if (row0 >= N) return;                 // wave-uniform: EXEC stays all-ones for WMMA
    const int half = lane >> 4;            // 0 or 1
    const int l16  = lane & 15;

    v8f c[8];
    const v8f zero = {};
    if (Cin) {
#pragma unroll
        for (int t = 0; t < 8; ++t)
#pragma unroll
            for (int g = 0; g < 8; ++g)
                c[t][g] = Cin[(size_t)(row0 + g + half * 8) * HD + t * 16 + l16];
    } else {
#pragma unroll
        for (int t = 0; t < 8; ++t) c[t] = zero;
    }

    const float* Arow = A + (size_t)(row0 + l16) * K;
    for (int k = 0; k < K; k += 4) {
        const v2f a = *(const v2f*)(Arow + k + half * 2);      // 8B-aligned (even idx)
        const float* Wk = W + (size_t)(k + half * 2) * HD;
#pragma unroll
        for (int t = 0; t < 8; ++t) {
            v2f b;
            b.x = Wk[t * 16 + l16];
            b.y = Wk[HD + t * 16 + l16];
            c[t] = __builtin_amdgcn_wmma_f32_16x16x4_f32(
                false, a, false, b, (short)0, c[t], false, false);
        }
    }

#pragma unroll
    for (int t = 0; t < 8; ++t) {
        const int col = t * 16 + l16;
        const float bv = bias ? bias[col] : 0.0f;
#pragma unroll
        for (int g = 0; g < 8; ++g) {
            float v = c[t][g] + bv;
            if (relu) v = fmaxf(v, 0.0f);
            Cout[(size_t)(row0 + g + half * 8) * HD + col] = v;
        }
    }
}

// -----------------------------------------------------------------------------
// Scatter-mean numerator + degree: one wave (32 lanes) per edge, 4 floats/lane.
// Node features are L2-resident (<=26 MB vs 192 MB L2), so gathers/atomics hit L2.
// -----------------------------------------------------------------------------
__global__ __launch_bounds__(256)
void aggregate(const float* __restrict__ hsrc,
               const int* __restrict__ sidx,
               const int* __restrict__ didx,
               int E,
               float* __restrict__ acc,
               float* __restrict__ deg)
{
    const int lane = threadIdx.x & 31;
    const int e = blockIdx.x * 8 + (threadIdx.x >> 5);
    if (e >= E) return;
    const int s = sidx[e];
    const int d = didx[e];
    const float4 v = *(const float4*)(hsrc + (size_t)s * HD + lane * 4);
    float* dst = acc + (size_t)d * HD + lane * 4;
    atomicAdd(dst + 0, v.x);
    atomicAdd(dst + 1, v.y);
    atomicAdd(dst + 2, v.z);
    atomicAdd(dst + 3, v.w);
    if (lane == 0) atomicAdd(deg + d, 1.0f);
}

__global__ __launch_bounds__(256)
void div_deg(float* __restrict__ acc, const float* __restrict__ deg, int total)
{
    const int i = blockIdx.x * blockDim.x + threadIdx.x;
    if (i < total) acc[i] /= fmaxf(deg[i >> 7], 1.0f);
}

// -----------------------------------------------------------------------------
// Classifier: out[e] = dot(h_pep[el_src[e]], h_prot[el_dst[e]]), 128-d, one wave/edge
// -----------------------------------------------------------------------------
__global__ __launch_bounds__(256)
void edge_dot(const float* __restrict__ hp, const float* __restrict__ hq,
              const int* __restrict__ es, const int* __restrict__ ed,
              int EL, float* __restrict__ out)
{
    const int lane = threadIdx.x & 31;
    const int e = blockIdx.x * 8 + (threadIdx.x >> 5);
    if (e >= EL) return;
    const int s = es[e];
    const int d = ed[e];
    const float4 a = *(const float4*)(hp + (size_t)s * HD + lane * 4);
    const float4 b = *(const float4*)(hq + (size_t)d * HD + lane * 4);
    float sum = a.x * b.x + a.y * b.y + a.z * b.z + a.w * b.w;
    for (int off = 16; off > 0; off >>= 1)
        sum += __shfl_xor(sum, off, 32);
    if (lane == 0) out[e] = sum;
}

extern "C" void kernel_launch(void* const* d_in, const int* in_sizes, int n_in,
                              void* d_out, int out_size, void* d_ws, size_t ws_size,
                              hipStream_t stream)
{
    (void)in_sizes; (void)n_in; (void)out_size; (void)ws_size;

    const float* x_pep    = (const float*)d_in[0];
    const float* x_prot   = (const float*)d_in[1];
    const int*   e_src    = (const int*)d_in[2];
    const int*   e_dst    = (const int*)d_in[3];
    const int*   el_src   = (const int*)d_in[4];
    const int*   el_dst   = (const int*)d_in[5];
    const float* W_pep    = (const float*)d_in[6];
    const float* b_pep    = (const float*)d_in[7];
    const float* W_prot   = (const float*)d_in[8];
    const float* b_prot   = (const float*)d_in[9];
    const float* Wl1_bind = (const float*)d_in[10];
    const float* Wr1_bind = (const float*)d_in[11];
    const float* Wl1_rev  = (const float*)d_in[12];
    const float* Wr1_rev  = (const float*)d_in[13];
    const float* Wl2_bind = (const float*)d_in[14];
    const float* Wr2_bind = (const float*)d_in[15];
    const float* Wl2_rev  = (const float*)d_in[16];
    const float* Wr2_rev  = (const float*)d_in[17];
    const float* b1_bind  = (const float*)d_in[18];
    const float* b1_rev   = (const float*)d_in[19];
    const float* b2_bind  = (const float*)d_in[20];
    const float* b2_rev   = (const float*)d_in[21];
    float* out = (float*)d_out;

    // scratch layout
    float* ws = (float*)d_ws;
    float* h_pep   = ws; ws += (size_t)NP_PEP  * HD;
    float* h_prot  = ws; ws += (size_t)NP_PROT * HD;
    float* h_pep2  = ws; ws += (size_t)NP_PEP  * HD;
    float* h_prot2 = ws; ws += (size_t)NP_PROT * HD;
    float* m_pep   = ws; ws += (size_t)NP_PEP  * HD;
    float* m_prot  = ws; ws += (size_t)NP_PROT * HD;
    float* deg_pep = ws; ws += NP_PEP;
    float* deg_prot= ws; ws += NP_PROT;

    auto gemm = [&](const float* A, int N, int K, const float* W,
                    const float* bias, const float* Cin, float* Cout, int relu) {
        const int blocks = (N / 16 + 7) / 8;
        gemm128_wmma<<<blocks, 256, 0, stream>>>(A, N, K, W, bias, Cin, Cout, relu);
    };
    auto mean_aggr = [&](const float* hsrc, const int* sidx, const int* didx,
                         float* acc, float* deg, int ndst) {
        hipMemsetAsync(acc, 0, (size_t)ndst * HD * sizeof(float), stream);
        hipMemsetAsync(deg, 0, (size_t)ndst * sizeof(float), stream);
        aggregate<<<(NE + 7) / 8, 256, 0, stream>>>(hsrc, sidx, didx, NE, acc, deg);
        const int total = ndst * HD;
        div_deg<<<(total + 255) / 256, 256, 0, stream>>>(acc, deg, total);
    };

    // input projections
    gemm(x_pep,  NP_PEP,  F_IN, W_pep,  b_pep,  nullptr, h_pep,  0);
    gemm(x_prot, NP_PROT, F_IN, W_prot, b_prot, nullptr, h_prot, 0);

    // SAGE layer 1 (+ReLU)
    mean_aggr(h_pep,  e_src, e_dst, m_prot, deg_prot, NP_PROT);   // pep->prot
    mean_aggr(h_prot, e_dst, e_src, m_pep,  deg_pep,  NP_PEP);    // prot->pep
    gemm(m_prot, NP_PROT, HD, Wl1_bind, b1_bind, nullptr, h_prot2, 0);
    gemm(h_prot, NP_PROT, HD, Wr1_bind, nullptr, h_prot2, h_prot2, 1);
    gemm(m_pep,  NP_PEP,  HD, Wl1_rev,  b1_rev,  nullptr, h_pep2,  0);
    gemm(h_pep,  NP_PEP,  HD, Wr1_rev,  nullptr, h_pep2,  h_pep2,  1);

    // SAGE layer 2 (no ReLU) — write back into h_pep / h_prot
    mean_aggr(h_pep2,  e_src, e_dst, m_prot, deg_prot, NP_PROT);
    mean_aggr(h_prot2, e_dst, e_src, m_pep,  deg_pep,  NP_PEP);
    gemm(m_prot,  NP_PROT, HD, Wl2_bind, b2_bind, nullptr, h_prot, 0);
    gemm(h_prot2, NP_PROT, HD, Wr2_bind, nullptr, h_prot,  h_prot, 0);
    gemm(m_pep,   NP_PEP,  HD, Wl2_rev,  b2_rev,  nullptr, h_pep,  0);
    gemm(h_pep2,  NP_PEP,  HD, Wr2_rev,  nullptr, h_pep,   h_pep,  0);

    // classifier
    edge_dot<<<(NEL + 7) / 8, 256, 0, stream>>>(h_pep, h_prot, el_src, el_dst, NEL, out);
}